// GCN_86973087744670
// MI455X (gfx1250) — compile-verified
//
#include <hip/hip_runtime.h>
#include <hip/hip_bf16.h>

typedef __attribute__((ext_vector_type(2))) float v2f;
typedef __attribute__((ext_vector_type(8))) float v8f;

#define N_NODES 50000
#define N_EDGES 800000
#define F 64

// ---------------- Kernel 1: zero the aggregation workspace ----------------
__global__ void GCN_zero_kernel(float* __restrict__ agg, int n) {
  int i = blockIdx.x * blockDim.x + threadIdx.x;
  if (i < n) agg[i] = 0.0f;
}

// ---------------- Kernel 2: edge scatter-add (segment_sum) ----------------
// One thread per (edge, feature). x row reads are coalesced; atomics land in
// L2 (agg is 12.8 MB, fully L2-resident on MI455X's 192 MB L2).
__global__ void GCN_scatter_kernel(const float* __restrict__ x,
                                   const int* __restrict__ edge_index,
                                   float* __restrict__ agg) {
  long long t = (long long)blockIdx.x * blockDim.x + threadIdx.x;
  int e = (int)(t >> 6);   // edge id
  int f = (int)(t & 63);   // feature id
  if (e >= N_EDGES) return;
  int s = edge_index[e];            // src = edge_index[0][e]
  int d = edge_index[N_EDGES + e];  // dst = edge_index[1][e]
  float v = x[(long long)s * F + f];
  atomicAdd(&agg[(long long)d * F + f], v);
}

// ---------------- Kernel 3: fused dual-GEMM + bias + ReLU via FP32 WMMA ---
// out = relu(agg @ W_rel^T + b_rel + x @ W_root^T)
// One wave per 16x16 output tile; 4 waves/block cover the 64 output features
// of one 16-row block. K=64 -> 16 V_WMMA_F32_16X16X4_F32 per GEMM.
__global__ __launch_bounds__(128) void GCN_gemm_kernel(
    const float* __restrict__ agg, const float* __restrict__ x,
    const float* __restrict__ W_rel, const float* __restrict__ b_rel,
    const float* __restrict__ W_root, float* __restrict__ out) {
  const int m0   = blockIdx.x * 16;          // output row tile
  const int wid  = threadIdx.x >> 5;         // wave id 0..3 -> N tile
  const int lane = threadIdx.x & 31;
  const int n0   = wid * 16;
  const int mr   = lane & 15;                // A: row in tile / B: col in tile
  const int kg   = (lane >> 4) * 2;          // K sub-offset within K=4 chunk

  // A layout (16x4 f32): lane<16 holds K=k+0,k+1 of row mr; lane>=16 holds K=k+2,k+3
  const float* aRow     = agg    + (long long)(m0 + mr) * F + kg;
  const float* xRow     = x      + (long long)(m0 + mr) * F + kg;
  // B[k][n] = W[n][k] (W stored [F_OUT, F_IN] row-major) -> contiguous float2 per lane
  const float* wRelRow  = W_rel  + (long long)(n0 + mr) * F + kg;
  const float* wRootRow = W_root + (long long)(n0 + mr) * F + kg;

  v8f acc = {};
#pragma unroll
  for (int k = 0; k < F; k += 4) {
    v2f a = *(const v2f*)(aRow + k);
    v2f b = *(const v2f*)(wRelRow + k);
    acc = __builtin_amdgcn_wmma_f32_16x16x4_f32(false, a, false, b,
                                                (short)0, acc, false, false);
  }
#pragma unroll
  for (int k = 0; k < F; k += 4) {
    v2f a = *(const v2f*)(xRow + k);
    v2f b = *(const v2f*)(wRootRow + k);
    acc = __builtin_amdgcn_wmma_f32_16x16x4_f32(false, a, false, b,
                                                (short)0, acc, false, false);
  }

  // C/D layout: VGPR r -> row (r + 8*(lane/16)), col (lane%16)
  const int col = n0 + mr;
  const int rowBase = m0 + ((lane >> 4) << 3);
  const float bias = b_rel[col];
#pragma unroll
  for (int r = 0; r < 8; ++r) {
    float v = acc[r] + bias;
    v = v > 0.0f ? v : 0.0f;
    out[(long long)(rowBase + r) * F + col] = v;
  }
}

extern "C" void kernel_launch(void* const* d_in, const int* in_sizes, int n_in,
                              void* d_out, int out_size, void* d_ws, size_t ws_size,
                              hipStream_t stream) {
  const float* x          = (const float*)d_in[0];
  const float* W_rel      = (const float*)d_in[1];
  const float* b_rel      = (const float*)d_in[2];
  const float* W_root     = (const float*)d_in[3];
  const int*   edge_index = (const int*)d_in[4];
  float* out = (float*)d_out;
  float* agg = (float*)d_ws;  // 50000*64*4 = 12.8 MB scratch

  const int aggN = N_NODES * F;
  GCN_zero_kernel<<<(aggN + 255) / 256, 256, 0, stream>>>(agg, aggN);

  const long long scatterThreads = (long long)N_EDGES * F;
  GCN_scatter_kernel<<<(int)((scatterThreads + 255) / 256), 256, 0, stream>>>(
      x, edge_index, agg);

  GCN_gemm_kernel<<<N_NODES / 16, 128, 0, stream>>>(agg, x, W_rel, b_rel,
                                                    W_root, out);
}